// ConsumptionPredictor_37606733644577
// MI455X (gfx1250) — compile-verified
//
#include <hip/hip_runtime.h>
#include <hip/hip_bf16.h>

#define HID 5
#define GATES 20          // 4*HID
#define SEQ 512
#define BATCH 16384
#define D1 512
#define KDIM (SEQ * HID)  // 2560

typedef __attribute__((ext_vector_type(2))) float v2f;
typedef __attribute__((ext_vector_type(8))) float v8f;

// ---------- fast activations (v_exp_f32 + v_rcp_f32 on the TRANS pipe) ----------
__device__ __forceinline__ float sigmoid_f(float x) {
    return __builtin_amdgcn_rcpf(1.0f + __expf(-x));
}
__device__ __forceinline__ float tanh_f(float x) {
    // tanh(x) = 2*sigmoid(2x) - 1
    return __builtin_fmaf(2.0f, __builtin_amdgcn_rcpf(1.0f + __expf(-2.0f * x)), -1.0f);
}

// ---------------------------------------------------------------------------
// Workspace layout (floats):
//   ws[0 .. 2559]   : wc = W2 @ W1            (1 x 2560)
//   ws[2560]        : c0 = W2 . b1 + b2       (scalar)
//   ws[2561 .. 2580]: b0sum = bih0 + bhh0     (20)
//   ws[2581 .. 2600]: b1sum = bih1 + bhh1     (20)
// ---------------------------------------------------------------------------

__global__ void prep_kernel(const float* __restrict__ bih0, const float* __restrict__ bhh0,
                            const float* __restrict__ bih1, const float* __restrict__ bhh1,
                            const float* __restrict__ W2,  const float* __restrict__ b1,
                            const float* __restrict__ b2,  float* __restrict__ ws) {
    int t = threadIdx.x;
    if (t < GATES) {
        ws[2561 + t] = bih0[t] + bhh0[t];
    } else if (t >= 32 && t < 32 + GATES) {
        int j = t - 32;
        ws[2581 + j] = bih1[j] + bhh1[j];
    } else if (t == 63) {
        float acc = b2[0];
        for (int j = 0; j < D1; ++j) acc = __builtin_fmaf(W2[j], b1[j], acc);
        ws[2560] = acc;
    }
}

// wc = W2 (1x512) @ W1 (512x2560), computed with V_WMMA_F32_16X16X4_F32.
// A (16x4) has all 16 rows replicated with W2[k..k+3]; only row M=0 of D is stored.
// One wave per 16-wide output tile; 2560/16 = 160 waves.
__global__ __launch_bounds__(32) void wc_wmma_kernel(const float* __restrict__ W1,
                                                     const float* __restrict__ W2,
                                                     float* __restrict__ ws) {
    const int n0   = blockIdx.x * 16;     // output column tile
    const int lane = threadIdx.x;         // 0..31
    const int half = lane >> 4;           // 0 -> K pair {0,1}, 1 -> K pair {2,3}
    const int l16  = lane & 15;           // N within tile (B/C/D) or M (A)

    v8f acc = {};
    for (int k0 = 0; k0 < D1; k0 += 4) {
        // A-matrix 16x4 f32: lanes 0-15 hold K=0,1 in v[0],v[1]; lanes 16-31 hold K=2,3.
        v2f a;
        a.x = W2[k0 + 2 * half + 0];
        a.y = W2[k0 + 2 * half + 1];
        // B-matrix 4x16 f32: lanes 0-15 hold rows K=0,1; lanes 16-31 rows K=2,3; N = lane%16.
        v2f b;
        b.x = W1[(size_t)(k0 + 2 * half + 0) * KDIM + n0 + l16];
        b.y = W1[(size_t)(k0 + 2 * half + 1) * KDIM + n0 + l16];
        acc = __builtin_amdgcn_wmma_f32_16x16x4_f32(
            /*neg_a=*/false, a, /*neg_b=*/false, b,
            /*c_mod=*/(short)0, acc, /*reuse_a=*/false, /*reuse_b=*/false);
    }
    // D layout: VGPR0, lanes 0-15 => M=0, N=lane
    if (lane < 16) ws[n0 + lane] = acc[0];
}

// Fully fused: 2-layer LSTM + collapsed MLP dot-product, one thread per batch row.
// All state (h1,c1,h2,c2) in registers; weights via uniform (scalar-cache) loads.
__global__ __launch_bounds__(32) void lstm_fused_kernel(
    const float* __restrict__ x,
    const float* __restrict__ Wih0, const float* __restrict__ Whh0,
    const float* __restrict__ Wih1, const float* __restrict__ Whh1,
    const float* __restrict__ ws,   float* __restrict__ out) {

    const int b = blockIdx.x * 32 + threadIdx.x;
    const float* __restrict__ xb = x + (size_t)b * SEQ;
    const float* __restrict__ wc = ws;            // 2560
    const float* __restrict__ b0 = ws + 2561;     // 20
    const float* __restrict__ b1 = ws + 2581;     // 20

    float h1[HID] = {}, c1[HID] = {}, h2[HID] = {}, c2[HID] = {};
    float acc = ws[2560];                         // c0 = W2.b1 + b2

    for (int s0 = 0; s0 < SEQ; s0 += 4) {
        const float4 xv4 = *(const float4*)(xb + s0);
        const float xs[4] = {xv4.x, xv4.y, xv4.z, xv4.w};

#pragma unroll
        for (int q = 0; q < 4; ++q) {
            const int s = s0 + q;
            const float xv = xs[q];
            float g[GATES];

            // ---- layer 0 gates: Wih0*x + b0 + Whh0 @ h1 ----
#pragma unroll
            for (int j = 0; j < GATES; ++j) {
                float a = __builtin_fmaf(Wih0[j], xv, b0[j]);
#pragma unroll
                for (int k = 0; k < HID; ++k)
                    a = __builtin_fmaf(Whh0[j * HID + k], h1[k], a);
                g[j] = a;
            }
#pragma unroll
            for (int k = 0; k < HID; ++k) {
                const float ig = sigmoid_f(g[k]);
                const float fg = sigmoid_f(g[HID + k]);
                const float gg = tanh_f(g[2 * HID + k]);
                const float og = sigmoid_f(g[3 * HID + k]);
                c1[k] = __builtin_fmaf(fg, c1[k], ig * gg);
                h1[k] = og * tanh_f(c1[k]);
            }

            // ---- layer 1 gates: Wih1 @ h1 + b1 + Whh1 @ h2 ----
#pragma unroll
            for (int j = 0; j < GATES; ++j) {
                float a = b1[j];
#pragma unroll
                for (int k = 0; k < HID; ++k)
                    a = __builtin_fmaf(Wih1[j * HID + k], h1[k], a);
#pragma unroll
                for (int k = 0; k < HID; ++k)
                    a = __builtin_fmaf(Whh1[j * HID + k], h2[k], a);
                g[j] = a;
            }
#pragma unroll
            for (int k = 0; k < HID; ++k) {
                const float ig = sigmoid_f(g[k]);
                const float fg = sigmoid_f(g[HID + k]);
                const float gg = tanh_f(g[2 * HID + k]);
                const float og = sigmoid_f(g[3 * HID + k]);
                c2[k] = __builtin_fmaf(fg, c2[k], ig * gg);
                h2[k] = og * tanh_f(c2[k]);
                // collapsed MLP: out += h2 . wc[s*5 + k]   (wc index is wave-uniform)
                acc = __builtin_fmaf(h2[k], wc[s * HID + k], acc);
            }
        }
    }
    out[b] = acc;
}

extern "C" void kernel_launch(void* const* d_in, const int* in_sizes, int n_in,
                              void* d_out, int out_size, void* d_ws, size_t ws_size,
                              hipStream_t stream) {
    const float* x    = (const float*)d_in[0];
    const float* Wih0 = (const float*)d_in[1];
    const float* Whh0 = (const float*)d_in[2];
    const float* bih0 = (const float*)d_in[3];
    const float* bhh0 = (const float*)d_in[4];
    const float* Wih1 = (const float*)d_in[5];
    const float* Whh1 = (const float*)d_in[6];
    const float* bih1 = (const float*)d_in[7];
    const float* bhh1 = (const float*)d_in[8];
    const float* W1   = (const float*)d_in[9];
    const float* b1   = (const float*)d_in[10];
    const float* W2   = (const float*)d_in[11];
    const float* b2   = (const float*)d_in[12];
    float* ws  = (float*)d_ws;
    float* out = (float*)d_out;

    prep_kernel<<<1, 64, 0, stream>>>(bih0, bhh0, bih1, bhh1, W2, b1, b2, ws);
    wc_wmma_kernel<<<KDIM / 16, 32, 0, stream>>>(W1, W2, ws);
    lstm_fused_kernel<<<BATCH / 32, 32, 0, stream>>>(x, Wih0, Whh0, Wih1, Whh1, ws, out);
}